// wFMConv_42554535969375
// MI455X (gfx1250) — compile-verified
//
#include <hip/hip_runtime.h>

// -------- CDNA5 async global->LDS path (guarded; fallback = sync copy) ------
#if __has_builtin(__builtin_amdgcn_global_load_async_to_lds_b64)
#define USE_ASYNC_LDS 1
#endif

// b64 async builtin takes pointers to v2i32 in AS1 (global) / AS3 (LDS).
typedef int v2i_t __attribute__((vector_size(8)));
typedef __attribute__((address_space(1))) v2i_t g_v2i;
typedef __attribute__((address_space(3))) v2i_t l_v2i;

__device__ __forceinline__ void async_copy_f2(const float2* __restrict__ g,
                                              float2* __restrict__ l) {
#ifdef USE_ASYNC_LDS
  __builtin_amdgcn_global_load_async_to_lds_b64(
      (g_v2i*)(void*)const_cast<float2*>(g),
      (l_v2i*)(void*)l, /*offset=*/0, /*cpol=*/0);
#else
  *l = *g;
#endif
}

template <int N>
__device__ __forceinline__ void wait_asynccnt() {
#ifdef USE_ASYNC_LDS
#if __has_builtin(__builtin_amdgcn_s_wait_asynccnt)
  __builtin_amdgcn_s_wait_asynccnt(N);
#else
  asm volatile("s_wait_asynccnt %0" ::"i"(N) : "memory");
#endif
#endif
}

// ---------------------------------------------------------------------------
#define TPB 256         // threads per block = batch elements per block (8 waves)
#define CHUNK 8         // k-steps staged per pipeline stage
#define ROWSTRIDE 9     // float2 units; odd -> conflict-free b64 LDS reads
#define KMAX 128

// One weighted-Frechet geodesic update on 2-vectors, mirroring the reference
// arithmetic EXACTLY — including the broadcast "S = 1.0 + s_i*s_j" (the 1.0
// is added to every element of the 2x2, not just the diagonal).
__device__ __forceinline__ void geo_step(float& m0, float& m1, float y0,
                                         float y1, float tw) {
  // (X·D)_j = (X·Y) X_j + (X·X) Y_j  with D_ij = Y_i X_j + X_i Y_j
  float xy = m0 * y0 + m1 * y1;
  float xx = m0 * m0 + m1 * m1;
  float l0 = y0 - 0.5f * (xy * m0 + xx * y0);
  float l1 = y1 - 0.5f * (xy * m1 + xx * y1);
  float s0 = tw * l0;
  float s1 = tw * l1;
  // S_ij = 1 + s_i * s_j   (broadcast add, per reference)
  float S00 = 1.0f + s0 * s0;
  float S01 = 1.0f + s0 * s1;  // == S10
  float S11 = 1.0f + s1 * s1;
  float det = S00 * S11 - S01 * S01;
  float u0 = m0 + s0;
  float u1 = m1 + s1;
  // result_j = u_i * adj(S)_ij / det
  m0 = (u0 * S11 - u1 * S01) / det;
  m1 = (u1 * S00 - u0 * S01) / det;
}

// Stage CHUNK k-steps for all TPB batch rows of this block into an LDS tile.
// Lane mapping: lk = t&7 walks k (contiguous 64B in global per 8 lanes),
// rows advance by 32 per pass -> coalesced async b64 stream.
__device__ __forceinline__ void stage_chunk(const float2* __restrict__ xg,
                                            float2* __restrict__ lbuf, int b0,
                                            int B, int K, int kbase, int lk,
                                            int lrow0) {
#pragma unroll
  for (int p = 0; p < CHUNK; ++p) {
    int row = lrow0 + p * (TPB / CHUNK);  // 0..TPB-1
    int bb = b0 + row;
    if (bb >= B) bb = B - 1;  // clamp instead of diverging
    const float2* g = xg + (size_t)bb * K + (kbase + lk);
    float2* l = lbuf + row * ROWSTRIDE + lk;
    async_copy_f2(g, l);
  }
}

__global__ __launch_bounds__(TPB) void wFMConv_kernel(
    const float2* __restrict__ x,   // (B, K) float2
    const float* __restrict__ kern, // (K)
    float2* __restrict__ out,       // (B)
    int B, int K) {
  __shared__ float2 tile[2][TPB * ROWSTRIDE];  // 36.9 KB double buffer
  __shared__ float wlds[KMAX];

  const int t = threadIdx.x;
  const int b0 = blockIdx.x * TPB;
  const int b = b0 + t;
  const int nchunks = K / CHUNK;

  if (t < K && t < KMAX) wlds[t] = kern[t];

  const int lk = t & (CHUNK - 1);
  const int lrow0 = t >> 3;  // CHUNK == 8

  // Prime the pipeline: chunk 0 -> buffer 0.
  stage_chunk(x, tile[0], b0, B, K, 0, lk, lrow0);

  float m0 = 0.0f, m1 = 0.0f;

  for (int c = 0; c < nchunks; ++c) {
    if (c + 1 < nchunks) {
      // Prefetch next chunk into the other buffer (its readers finished at
      // the trailing barrier of iteration c-1).
      stage_chunk(x, tile[(c + 1) & 1], b0, B, K, (c + 1) * CHUNK, lk, lrow0);
      wait_asynccnt<CHUNK>();  // chunk c's asyncs (older, in-order) are done
    } else {
      wait_asynccnt<0>();
    }
    __syncthreads();  // chunk c visible from every wave's DMA

    const float2* rowp = tile[c & 1] + t * ROWSTRIDE;
    if (c == 0) {
      float2 p = rowp[0];  // initial point X = x[b,0]
      m0 = p.x;
      m1 = p.y;
      float2 q = rowp[1];  // first update uses w[0] (w[1] unused by ref)
      geo_step(m0, m1, q.x, q.y, wlds[0]);
#pragma unroll
      for (int kk = 2; kk < CHUNK; ++kk) {
        float2 y = rowp[kk];
        geo_step(m0, m1, y.x, y.y, wlds[kk]);
      }
    } else {
      const int kbase = c * CHUNK;
#pragma unroll
      for (int kk = 0; kk < CHUNK; ++kk) {
        float2 y = rowp[kk];
        geo_step(m0, m1, y.x, y.y, wlds[kbase + kk]);
      }
    }
    __syncthreads();  // buffer (c&1) free for reuse by the next prefetch
  }

  if (b < B) out[b] = make_float2(m0, m1);
}

extern "C" void kernel_launch(void* const* d_in, const int* in_sizes, int n_in,
                              void* d_out, int out_size, void* d_ws,
                              size_t ws_size, hipStream_t stream) {
  (void)n_in;
  (void)d_ws;
  (void)ws_size;
  (void)out_size;
  const float* x = (const float*)d_in[0];     // (B, 128, 2) f32
  const float* kern = (const float*)d_in[1];  // (128, 1) f32
  float* out = (float*)d_out;                 // (B, 1, 2) f32

  const int K = in_sizes[1];                    // 128
  const long B = (long)in_sizes[0] / (2L * K);  // 131072
  const int grid = (int)((B + TPB - 1) / TPB);

  wFMConv_kernel<<<grid, TPB, 0, stream>>>((const float2*)x, kern, (float2*)out,
                                           (int)B, K);
}